// WinnogradRNN_61005715472776
// MI455X (gfx1250) — compile-verified
//
#include <hip/hip_runtime.h>
#include <hip/hip_bf16.h>

// Model dims
#define SQ   1024      // sequence length S
#define EMB  512       // embedding dim E
#define HID  512       // hidden H
#define TH3  1536      // 3*H
#define DD   1024      // dense dim D

typedef __attribute__((ext_vector_type(2))) float v2f;
typedef __attribute__((ext_vector_type(8))) float v8f;

#if __has_builtin(__builtin_amdgcn_wmma_f32_16x16x4_f32)
#define USE_WMMA_F32 1
#else
#define USE_WMMA_F32 0
#endif

__device__ __forceinline__ float sigm(float x) { return 1.0f / (1.0f + __expf(-x)); }

// ---------------------------------------------------------------------------
// Kernel 1: gi = X @ Wih^T + bih for one GRU.
// X[t] = emb[sent[t]] gathered on the fly. One wave computes one 16x16 tile
// of G[t][j] via V_WMMA_F32_16X16X4_F32 over K=512 (128 WMMAs).
// grid = (TH3/16, SQ/16), block = 32 (one wave; EXEC all-ones for WMMA).
// ---------------------------------------------------------------------------
__global__ __launch_bounds__(32) void embed_gemm(const int* __restrict__ sent,
                                                 const float* __restrict__ emb,
                                                 const float* __restrict__ Wih,
                                                 const float* __restrict__ bih,
                                                 float* __restrict__ G)
{
    const int n0   = blockIdx.x * 16;   // output gate-row tile (0..1535)
    const int t0   = blockIdx.y * 16;   // time tile
    const int lane = threadIdx.x;
    const int m    = lane & 15;
    const int hi   = lane >> 4;
    const int kk   = hi * 2;

    const float* xrow = emb + (long long)sent[t0 + m] * EMB;   // A: row m of X tile
    const float* wrow = Wih + (long long)(n0 + m) * EMB;       // B: col m of W^T tile

    v8f acc = {};

#if USE_WMMA_F32
    #pragma unroll 4
    for (int k = 0; k < EMB; k += 4) {
        v2f a, b;
        a.x = xrow[k + kk];  a.y = xrow[k + kk + 1];
        b.x = wrow[k + kk];  b.y = wrow[k + kk + 1];
        acc = __builtin_amdgcn_wmma_f32_16x16x4_f32(
                  /*neg_a=*/false, a, /*neg_b=*/false, b,
                  /*c_mod=*/(short)0, acc, /*reuse_a=*/false, /*reuse_b=*/false);
    }
#else
    // Scalar fallback with identical output mapping: acc[r] = G[t0+r+8*hi][n0+m]
    for (int r = 0; r < 8; ++r) {
        const float* xr = emb + (long long)sent[t0 + r + 8 * hi] * EMB;
        float s = 0.0f;
        for (int k = 0; k < EMB; ++k) s = fmaf(xr[k], wrow[k], s);
        acc[r] = s;
    }
#endif

    const float bias = bih[n0 + m];
    float* outp = G + (long long)t0 * TH3 + n0 + m;
    #pragma unroll
    for (int r = 0; r < 8; ++r)
        outp[(long long)(r + 8 * hi) * TH3] = acc[r] + bias;
}

// ---------------------------------------------------------------------------
// Kernel 2: transpose + gate-pack Whh [1536][512] -> Wt[k][j][{r,z,n}]
// so the scan's weight stream is coalesced (12B per lane, contiguous / wave).
// LDS-tiled 32x32. grid = (512/32, 1536/32), block = (32,8).
// ---------------------------------------------------------------------------
__global__ __launch_bounds__(256) void whh_transpose(const float* __restrict__ Whh,
                                                     float* __restrict__ Wt)
{
    __shared__ float tile[32][33];
    const int kt = blockIdx.x * 32;   // k tile
    const int rt = blockIdx.y * 32;   // row tile (rows 0..1535 span the 3 gates)
    const int tx = threadIdx.x;
    const int ty = threadIdx.y;

    #pragma unroll
    for (int i = 0; i < 32; i += 8)
        tile[ty + i][tx] = Whh[(long long)(rt + ty + i) * HID + kt + tx];
    __syncthreads();

    const int row = rt + tx;
    const int g   = row >> 9;     // gate index 0/1/2 (tiles never straddle: 512 % 32 == 0)
    const int j   = row & (HID - 1);
    #pragma unroll
    for (int i = 0; i < 32; i += 8) {
        const int k = kt + ty + i;
        Wt[((long long)k * HID + j) * 3 + g] = tile[tx][ty + i];
    }
}

// ---------------------------------------------------------------------------
// Kernel 3: the sequential GRU scan. 4 blocks (one per GRU), 512 threads.
// Thread j owns hidden unit j: per step, three 512-long dots against h (LDS,
// broadcast reads), then the gate math. Whh stream (3MB/step) stays in L2.
// ---------------------------------------------------------------------------
struct GruArgs {
    const float* Wt;    // [512][512][3]
    const float* bhh;   // [1536]
    const float* G;     // [1024][1536]
    float*       hout;  // [512]
    int          reverse;
};

__global__ __launch_bounds__(512) void gru_scan(GruArgs a0, GruArgs a1, GruArgs a2, GruArgs a3)
{
    GruArgs a = (blockIdx.x == 0) ? a0 : (blockIdx.x == 1) ? a1 : (blockIdx.x == 2) ? a2 : a3;

    __shared__ float h[HID];
    const int j = threadIdx.x;
    h[j] = 0.0f;
    const float bhr = a.bhh[j], bhz = a.bhh[HID + j], bhn = a.bhh[2 * HID + j];
    __syncthreads();

    for (int t = 0; t < SQ; ++t) {
        const int tt = a.reverse ? (SQ - 1 - t) : t;
        const float* gt = a.G + (long long)tt * TH3;

        if (t + 1 < SQ) {   // prefetch next step's gi row into cache
            const int tn = a.reverse ? (SQ - 2 - t) : (t + 1);
            const float* gn = a.G + (long long)tn * TH3;
            __builtin_prefetch(gn + j, 0, 1);
            __builtin_prefetch(gn + HID + j, 0, 1);
            __builtin_prefetch(gn + 2 * HID + j, 0, 1);
        }

        const float* w = a.Wt + j * 3;
        float ar = 0.0f, az = 0.0f, an = 0.0f;
        #pragma unroll 8
        for (int k = 0; k < HID; ++k) {
            const float hk = h[k];
            ar = fmaf(w[0], hk, ar);
            az = fmaf(w[1], hk, az);
            an = fmaf(w[2], hk, an);
            w += TH3;
        }

        const float r    = sigm(gt[j] + bhr + ar);
        const float z    = sigm(gt[HID + j] + bhz + az);
        const float n    = tanhf(gt[2 * HID + j] + r * (bhn + an));
        const float hnew = (1.0f - z) * n + z * h[j];

        __syncthreads();
        h[j] = hnew;
        __syncthreads();
    }
    a.hout[j] = h[j];
}

// ---------------------------------------------------------------------------
// Kernel 4: head. rep[2048] = concat of the 4 final hiddens (already laid out
// contiguously: ctx_f, ctx_b, qry_f, qry_b). h1 = relu(rep @ d1_w^T + d1_b),
// out = sigmoid(h1 . d2_w + d2_b). One block of 1024 threads + LDS reduce.
// ---------------------------------------------------------------------------
__global__ __launch_bounds__(1024) void dense_head(const float* __restrict__ rep,
                                                   const float* __restrict__ d1w,
                                                   const float* __restrict__ d1b,
                                                   const float* __restrict__ d2w,
                                                   const float* __restrict__ d2b,
                                                   float* __restrict__ outp)
{
    __shared__ float srep[4 * HID];
    __shared__ float red[DD];
    const int i = threadIdx.x;
    srep[i]        = rep[i];
    srep[i + 1024] = rep[i + 1024];
    __syncthreads();

    const float* wr = d1w + (long long)i * (4 * HID);
    float s = 0.0f;
    #pragma unroll 4
    for (int k = 0; k < 4 * HID; ++k) s = fmaf(wr[k], srep[k], s);
    float h1 = s + d1b[i];
    h1 = h1 > 0.0f ? h1 : 0.0f;
    red[i] = h1 * d2w[i];
    __syncthreads();

    for (int off = DD / 2; off > 0; off >>= 1) {
        if (i < off) red[i] += red[i + off];
        __syncthreads();
    }
    if (i == 0) outp[0] = sigm(red[0] + d2b[0]);
}

// ---------------------------------------------------------------------------
// Launcher. Input order (setup_inputs): sentence, emb,
// then 4x {Wih, Whh, bih, bhh} for ctx_f, ctx_b, qry_f, qry_b,
// then d1_w, d1_b, d2_w, d2_b.
// ---------------------------------------------------------------------------
extern "C" void kernel_launch(void* const* d_in, const int* in_sizes, int n_in,
                              void* d_out, int out_size, void* d_ws, size_t ws_size,
                              hipStream_t stream)
{
    (void)in_sizes; (void)n_in; (void)out_size; (void)ws_size;

    const int*   sent = (const int*)d_in[0];
    const float* emb  = (const float*)d_in[1];

    // Workspace layout (floats): WhhT[4][512*1536] | G[4][1024*1536] | hfin[4*512]
    float* ws   = (float*)d_ws;
    float* WhhT = ws;
    float* G    = WhhT + (size_t)4 * HID * TH3;        // + 3,145,728
    float* hfin = G    + (size_t)4 * SQ * TH3;         // + 6,291,456

    for (int g = 0; g < 4; ++g) {
        const float* Wih = (const float*)d_in[2 + 4 * g];
        const float* Whh = (const float*)d_in[3 + 4 * g];
        const float* bih = (const float*)d_in[4 + 4 * g];
        embed_gemm<<<dim3(TH3 / 16, SQ / 16), 32, 0, stream>>>(
            sent, emb, Wih, bih, G + (size_t)g * SQ * TH3);
        whh_transpose<<<dim3(HID / 32, TH3 / 32), dim3(32, 8), 0, stream>>>(
            Whh, WhhT + (size_t)g * HID * TH3);
    }

    GruArgs a[4];
    for (int g = 0; g < 4; ++g) {
        a[g].Wt      = WhhT + (size_t)g * HID * TH3;
        a[g].bhh     = (const float*)d_in[5 + 4 * g];
        a[g].G       = G + (size_t)g * SQ * TH3;
        a[g].hout    = hfin + g * HID;
        a[g].reverse = (g == 1 || g == 3);   // ctx_b, qry_b consume reversed sequence
    }
    gru_scan<<<4, HID, 0, stream>>>(a[0], a[1], a[2], a[3]);

    dense_head<<<1, DD, 0, stream>>>(hfin,
                                     (const float*)d_in[18], (const float*)d_in[19],
                                     (const float*)d_in[20], (const float*)d_in[21],
                                     (float*)d_out);
}